// TCVAE_37546604102034
// MI455X (gfx1250) — compile-verified
//
#include <hip/hip_runtime.h>
#include <hip/hip_bf16.h>
#include <math.h>

typedef __attribute__((ext_vector_type(2))) float v2f;
typedef __attribute__((ext_vector_type(4))) float v4f;
typedef __attribute__((ext_vector_type(8))) float v8f;

#define BATCH 4096
#define DIM   512
#define LAT   16
#define NPATH (LAT + 1)          // 16 per-latent LSE paths + 1 sum-over-latents path
#define JSEG  8                  // j-range split for occupancy (2048 waves total)
#define JT_PER_SEG (BATCH / 16 / JSEG)   // 32 tiles of 16 j's per segment
#define LOG_2PI 1.8378770664093453f

// ---- workspace layout (in floats) ----
// b1[l][j] = -0.5 * exp(-lsv[j,l])
// b2[l][j] = mean[j,l] * exp(-lsv[j,l])
// b3[l][j] = -0.5 * (mean^2*exp(-lsv) + lsv + LOG_2PI)[j,l]
#define WS_B1 0
#define WS_B2 (BATCH * LAT)
#define WS_B3 (2 * BATCH * LAT)
#define NRP 512
#define NKP 64
#define NTP 16
#define WS_RP (3 * BATCH * LAT)
#define WS_KP (WS_RP + NRP)
#define NPART (NPATH * BATCH * JSEG)     // 17 * 4096 * 8 = 557056
#define WS_MP (WS_KP + NKP)              // running-max partials
#define WS_SP (WS_MP + NPART)            // sum-exp partials
#define WS_TP (WS_SP + NPART)            // 16 tc partials after merge

// ---------------- prep: per-column Gaussian stats ----------------
__global__ void tc_prep_kernel(const float* __restrict__ zmean,
                               const float* __restrict__ zlogvar,
                               float* __restrict__ ws) {
    int idx = blockIdx.x * blockDim.x + threadIdx.x;   // 0 .. 65535
    int l = idx >> 12;          // latent
    int j = idx & (BATCH - 1);  // batch column
    float m = zmean[j * LAT + l];
    float v = zlogvar[j * LAT + l];
    float inv = __expf(-v);
    ws[WS_B1 + l * BATCH + j] = -0.5f * inv;
    ws[WS_B2 + l * BATCH + j] = m * inv;
    ws[WS_B3 + l * BATCH + j] = -0.5f * (m * m * inv + v + LOG_2PI);
}

// ---------------- recon: sum |a-b| ----------------
__global__ void recon_kernel(const float* __restrict__ a,
                             const float* __restrict__ b,
                             float* __restrict__ part) {
    __shared__ float sm[256];
    int tid = threadIdx.x;
    int gid = blockIdx.x * blockDim.x + tid;
    int stride = gridDim.x * blockDim.x;
    const v4f* a4 = (const v4f*)a;
    const v4f* b4 = (const v4f*)b;
    const int n4 = (BATCH * DIM) / 4;
    float s = 0.0f;
    for (int i = gid; i < n4; i += stride) {
        v4f x = a4[i];
        v4f y = b4[i];
        s += fabsf(x.x - y.x) + fabsf(x.y - y.y) +
             fabsf(x.z - y.z) + fabsf(x.w - y.w);
    }
    sm[tid] = s;
    __syncthreads();
    for (int o = 128; o > 0; o >>= 1) {
        if (tid < o) sm[tid] += sm[tid + o];
        __syncthreads();
    }
    if (tid == 0) part[blockIdx.x] = sm[0];
}

// ---------------- KL: sum (m^2 + e^v - v - 1) ----------------
__global__ void kl_kernel(const float* __restrict__ zmean,
                          const float* __restrict__ zlogvar,
                          float* __restrict__ part) {
    __shared__ float sm[256];
    int tid = threadIdx.x;
    int gid = blockIdx.x * blockDim.x + tid;
    int stride = gridDim.x * blockDim.x;
    float s = 0.0f;
    for (int i = gid; i < BATCH * LAT; i += stride) {
        float m = zmean[i];
        float v = zlogvar[i];
        s += m * m + __expf(v) - v - 1.0f;
    }
    sm[tid] = s;
    __syncthreads();
    for (int o = 128; o > 0; o >>= 1) {
        if (tid < o) sm[tid] += sm[tid + o];
        __syncthreads();
    }
    if (tid == 0) part[blockIdx.x] = sm[0];
}

// ---------------- TC: WMMA pairwise log-density + online logsumexp ----------------
// One wave per (16-i-block, j-segment). D-tile layout: M = j (reduce axis:
// in-lane over 8 VGPRs, then lane^16 shuffle), N = i (one per lane&15).
// Each wave emits partial (max, sumexp) per (i, path) for a later merge.
__global__ __launch_bounds__(32) void tc_kernel(const float* __restrict__ z,
                                                const float* __restrict__ ws,
                                                float* __restrict__ m_part,
                                                float* __restrict__ s_part) {
    const float* b1 = ws + WS_B1;
    const float* b2 = ws + WS_B2;
    const float* b3 = ws + WS_B3;

    const int lane = threadIdx.x;            // 0..31
    const int li   = lane & 15;
    const int hi   = lane >> 4;              // 0: K0/K1 half, 1: K2/K3 half
    const int ib   = blockIdx.x >> 3;        // i-block (0..255)
    const int seg  = blockIdx.x & (JSEG - 1);
    const int i    = ib * 16 + li;

    // B operand (4x16, K x N=i) per latent:
    //   K0 = z^2, K1 = z, K2 = 1, K3 = 0   (lanes 0-15 carry K0/K1; 16-31 K2/K3)
    v2f bmat[LAT];
#pragma unroll
    for (int l = 0; l < LAT; ++l) {
        float zz = z[i * LAT + l];
        bmat[l].x = hi ? 1.0f : zz * zz;
        bmat[l].y = hi ? 0.0f : zz;
    }

    // online logsumexp state: per latent + sum-over-latents path
    float mL[LAT], sL[LAT];
#pragma unroll
    for (int l = 0; l < LAT; ++l) { mL[l] = -3.0e38f; sL[l] = 0.0f; }
    float mS = -3.0e38f, sS = 0.0f;

    const int jt0 = seg * JT_PER_SEG;
    for (int jt = jt0; jt < jt0 + JT_PER_SEG; ++jt) {
        const int j = jt * 16 + li;   // this lane's column stat row

        float sacc[8];
#pragma unroll
        for (int r = 0; r < 8; ++r) sacc[r] = 0.0f;

#pragma unroll
        for (int l = 0; l < LAT; ++l) {
            // A operand (16x4, M=j x K): K0=b1, K1=b2, K2=b3, K3=0
            float v1 = b1[l * BATCH + j];
            float v2 = b2[l * BATCH + j];
            float v3 = b3[l * BATCH + j];
            v2f amat;
            amat.x = hi ? v3 : v1;
            amat.y = hi ? 0.0f : v2;

            v8f c = {0.f, 0.f, 0.f, 0.f, 0.f, 0.f, 0.f, 0.f};
            v8f d = __builtin_amdgcn_wmma_f32_16x16x4_f32(
                false, amat, false, bmat[l], (short)0, c, false, false);

            // this lane holds 8 j-rows of the tile for its i column
            float tmax = d[0];
#pragma unroll
            for (int r = 1; r < 8; ++r) tmax = fmaxf(tmax, d[r]);
            float mnew = fmaxf(mL[l], tmax);
            float acc = sL[l] * __expf(mL[l] - mnew);
#pragma unroll
            for (int r = 0; r < 8; ++r) acc += __expf(d[r] - mnew);
            sL[l] = acc;
            mL[l] = mnew;

#pragma unroll
            for (int r = 0; r < 8; ++r) sacc[r] += d[r];
        }

        // sum-over-latents path for this tile's 8 j-rows
        float tmax = sacc[0];
#pragma unroll
        for (int r = 1; r < 8; ++r) tmax = fmaxf(tmax, sacc[r]);
        float mnew = fmaxf(mS, tmax);
        float acc = sS * __expf(mS - mnew);
#pragma unroll
        for (int r = 0; r < 8; ++r) acc += __expf(sacc[r] - mnew);
        sS = acc;
        mS = mnew;
    }

    // combine the two half-wave j partitions (lane <-> lane^16), emit partials
#pragma unroll
    for (int l = 0; l < LAT; ++l) {
        float mo = __shfl_xor(mL[l], 16, 32);
        float so = __shfl_xor(sL[l], 16, 32);
        float M = fmaxf(mL[l], mo);
        float s = sL[l] * __expf(mL[l] - M) + so * __expf(mo - M);
        if (lane < 16) {
            m_part[(l * BATCH + i) * JSEG + seg] = M;
            s_part[(l * BATCH + i) * JSEG + seg] = s;
        }
    }
    float mo = __shfl_xor(mS, 16, 32);
    float so = __shfl_xor(sS, 16, 32);
    float M = fmaxf(mS, mo);
    float s = sS * __expf(mS - M) + so * __expf(mo - M);
    if (lane < 16) {
        m_part[(LAT * BATCH + i) * JSEG + seg] = M;
        s_part[(LAT * BATCH + i) * JSEG + seg] = s;
    }
}

// ---------------- merge segments, form tc_i, block-reduce ----------------
__global__ void tc_merge_kernel(const float* __restrict__ m_part,
                                const float* __restrict__ s_part,
                                float* __restrict__ part) {
    __shared__ float sm[256];
    int tid = threadIdx.x;
    int i = blockIdx.x * 256 + tid;   // 0..4095

    float lse[NPATH];
#pragma unroll
    for (int p = 0; p < NPATH; ++p) {
        float M = -3.0e38f;
#pragma unroll
        for (int k = 0; k < JSEG; ++k)
            M = fmaxf(M, m_part[(p * BATCH + i) * JSEG + k]);
        float s = 0.0f;
#pragma unroll
        for (int k = 0; k < JSEG; ++k)
            s += s_part[(p * BATCH + i) * JSEG + k] *
                 __expf(m_part[(p * BATCH + i) * JSEG + k] - M);
        lse[p] = M + __logf(s);
    }
    float log_qz_product = 0.0f;
#pragma unroll
    for (int l = 0; l < LAT; ++l) log_qz_product += lse[l];
    float tc = lse[LAT] - log_qz_product;

    sm[tid] = tc;
    __syncthreads();
    for (int o = 128; o > 0; o >>= 1) {
        if (tid < o) sm[tid] += sm[tid + o];
        __syncthreads();
    }
    if (tid == 0) part[blockIdx.x] = sm[0];
}

// ---------------- final combine ----------------
__global__ void final_kernel(const float* __restrict__ ws, float* __restrict__ out) {
    __shared__ float sr[256], sk[256], st[256];
    int tid = threadIdx.x;
    float ar = 0.f, ak = 0.f, at = 0.f;
    for (int i = tid; i < NRP; i += 256) ar += ws[WS_RP + i];
    for (int i = tid; i < NKP; i += 256) ak += ws[WS_KP + i];
    for (int i = tid; i < NTP; i += 256) at += ws[WS_TP + i];
    sr[tid] = ar; sk[tid] = ak; st[tid] = at;
    __syncthreads();
    for (int o = 128; o > 0; o >>= 1) {
        if (tid < o) { sr[tid] += sr[tid + o]; sk[tid] += sk[tid + o]; st[tid] += st[tid + o]; }
        __syncthreads();
    }
    if (tid == 0) {
        float recon = sr[0] / (float)(BATCH * DIM);
        float kl    = 0.5f * sk[0] / (float)BATCH;
        float tc    = st[0] / (float)BATCH;
        out[0] = recon + kl + tc;
    }
}

extern "C" void kernel_launch(void* const* d_in, const int* in_sizes, int n_in,
                              void* d_out, int out_size, void* d_ws, size_t ws_size,
                              hipStream_t stream) {
    const float* data   = (const float*)d_in[0];
    const float* recon  = (const float*)d_in[1];
    const float* z      = (const float*)d_in[2];
    const float* zmean  = (const float*)d_in[3];
    const float* zlv    = (const float*)d_in[4];
    float* ws  = (float*)d_ws;
    float* out = (float*)d_out;

    tc_prep_kernel<<<BATCH * LAT / 256, 256, 0, stream>>>(zmean, zlv, ws);
    recon_kernel<<<NRP, 256, 0, stream>>>(data, recon, ws + WS_RP);
    kl_kernel<<<NKP, 256, 0, stream>>>(zmean, zlv, ws + WS_KP);
    tc_kernel<<<256 * JSEG, 32, 0, stream>>>(z, ws, ws + WS_MP, ws + WS_SP);
    tc_merge_kernel<<<BATCH / 256, 256, 0, stream>>>(ws + WS_MP, ws + WS_SP, ws + WS_TP);
    final_kernel<<<1, 256, 0, stream>>>(ws, out);
}